// AllToAllDispatchBackward_15822659519277
// MI455X (gfx1250) — compile-verified
//
#include <hip/hip_runtime.h>

// ---------------------------------------------------------------------------
// AllToAllDispatch: out[d, t, :] = mask(d, t) * in[t, :]
//   T=8192, H=4096, E=64, K=2, D=8.
// Memory-bound: 1 GiB NT writes + 128 MiB reads => ~48 us roofline @23.3TB/s.
//
// Data rows (~22% of writes): input row held in VGPRs, direct NT b128 stores.
// Zero rows (~78% of writes): streamed from a zeroed 16KB LDS buffer via the
// CDNA5 async store path (global_store_async_from_lds_b128, ASYNCcnt) -- the
// constant bytes never touch VGPRs and overlap with the data-row stores.
// ---------------------------------------------------------------------------

#define NUM_DEV 8
#define T_TOK   8192
#define H_DIM   4096
#define K_TOP   2

typedef float v4f __attribute__((ext_vector_type(4)));

__global__ __launch_bounds__(256) void a2a_dispatch_kernel(
    const float* __restrict__ in,     // [T, H]
    const int*   __restrict__ eidx,   // [T, K]
    const int*   __restrict__ emap,   // [E]
    float*       __restrict__ out)    // [D, T, H]
{
    const int t   = blockIdx.x;
    const int tid = threadIdx.x;

    // 16 KB of zeros in LDS; source operand for async zero-row stores.
    __shared__ v4f zrow[H_DIM / 4];

    // Wave-uniform routing lookup -> scalar loads.
    const int e0 = eidx[t * K_TOP + 0];
    const int e1 = eidx[t * K_TOP + 1];
    const int d0 = emap[e0];
    const int d1 = emap[e1];

    // Stage this thread's slice of the row in registers (read HBM once).
    const v4f* __restrict__ rin = (const v4f*)(in + (size_t)t * H_DIM);
    v4f vals[4];
#pragma unroll
    for (int j = 0; j < 4; ++j) {
        vals[j] = __builtin_nontemporal_load(&rin[tid + 256 * j]);
    }

    // Zero-init the LDS staging buffer (ds_store_b128), then barrier before
    // any async store reads it.
    const v4f zero = {0.f, 0.f, 0.f, 0.f};
#pragma unroll
    for (int j = 0; j < 4; ++j) {
        zrow[tid + 256 * j] = zero;
    }
    __syncthreads();

    // Low 32 bits of the flat address of a __shared__ object are the
    // wave-relative LDS byte offset.
    const unsigned lds_base = (unsigned)(uintptr_t)(void*)&zrow[0];

#pragma unroll
    for (int d = 0; d < NUM_DEV; ++d) {
        v4f* __restrict__ rout =
            (v4f*)(out + ((size_t)d * T_TOK + (size_t)t) * H_DIM);
        const bool on = (d == d0) | (d == d1);   // uniform per block
        if (on) {
            // Data row: direct non-temporal b128 stores from registers.
#pragma unroll
            for (int j = 0; j < 4; ++j) {
                __builtin_nontemporal_store(vals[j], &rout[tid + 256 * j]);
            }
        } else {
            // Zero row: CDNA5 async LDS->global streaming stores (ASYNCcnt).
#pragma unroll
            for (int j = 0; j < 4; ++j) {
                unsigned long long gaddr =
                    (unsigned long long)(uintptr_t)&rout[tid + 256 * j];
                unsigned lds_off = lds_base + 16u * (unsigned)(tid + 256 * j);
                asm volatile(
                    "global_store_async_from_lds_b128 %0, %1, off th:TH_STORE_NT"
                    :
                    : "v"(gaddr), "v"(lds_off)
                    : "memory");
            }
        }
    }

    // Retire all outstanding async zero-row stores before wave end.
#if __has_builtin(__builtin_amdgcn_s_wait_asynccnt)
    __builtin_amdgcn_s_wait_asynccnt(0);
#else
    asm volatile("s_wait_asynccnt 0x0" ::: "memory");
#endif
}

extern "C" void kernel_launch(void* const* d_in, const int* in_sizes, int n_in,
                              void* d_out, int out_size, void* d_ws, size_t ws_size,
                              hipStream_t stream) {
    const float* input = (const float*)d_in[0];   // [T, H] float32
    const int*   eidx  = (const int*)d_in[1];     // [T, K] int32
    const int*   emap  = (const int*)d_in[2];     // [E]    int32
    float*       out   = (float*)d_out;           // [D, T, H] float32

    (void)in_sizes; (void)n_in; (void)out_size; (void)d_ws; (void)ws_size;

    dim3 grid(T_TOK);
    dim3 block(256);
    hipLaunchKernelGGL(a2a_dispatch_kernel, grid, block, 0, stream,
                       input, eidx, emap, out);
}